// SSMMoETSP_26757646254307
// MI455X (gfx1250) — compile-verified
//
#include <hip/hip_runtime.h>
#include <hip/hip_bf16.h>

typedef float v2f __attribute__((ext_vector_type(2)));
typedef float v8f __attribute__((ext_vector_type(8)));

#define Lc 3
#define Ec 8
#define Bc 256
#define Nc 1024
#define Dc 512
#define DSc 128

// ---------------- init: h = token, lb_acc = 0 ----------------
__global__ void init_kernel(const float* __restrict__ token, float* __restrict__ h,
                            float* __restrict__ lb_acc) {
    int idx = blockIdx.x * blockDim.x + threadIdx.x;
    if (idx < Bc * Dc) h[idx] = token[idx];
    if (idx == 0) *lb_acc = 0.0f;
}

// ---------------- gating: probs/gate/sel per (b) ----------------
__global__ void gate_kernel(const float* __restrict__ h,
                            const float* __restrict__ Wg_l,  // (D,E)
                            const float* __restrict__ bg_l,  // (E)
                            float* __restrict__ probs,       // (B,E)
                            float* __restrict__ gate,        // (B,E)
                            float* __restrict__ sel) {       // (B,E)
    __shared__ float lg[Ec];
    const int b = blockIdx.x;
    const int lane = threadIdx.x & 31;
    const int e = threadIdx.x >> 5;   // 8 waves -> 8 experts
    float s = 0.0f;
    for (int d = lane; d < Dc; d += 32)
        s += h[b * Dc + d] * Wg_l[d * Ec + e];
    for (int off = 16; off; off >>= 1) s += __shfl_xor(s, off, 32);
    if (lane == 0) lg[e] = s + bg_l[e];
    __syncthreads();
    if (threadIdx.x == 0) {
        float mx = lg[0];
        for (int i = 1; i < Ec; ++i) mx = fmaxf(mx, lg[i]);
        float p[Ec]; float sum = 0.0f;
        for (int i = 0; i < Ec; ++i) { p[i] = __expf(lg[i] - mx); sum += p[i]; }
        for (int i = 0; i < Ec; ++i) p[i] /= sum;
        // top-2, earliest index on ties (matches jax.lax.top_k)
        int i1 = 0;
        for (int i = 1; i < Ec; ++i) if (p[i] > p[i1]) i1 = i;
        int i2 = (i1 == 0) ? 1 : 0;
        for (int i = 0; i < Ec; ++i) if (i != i1 && p[i] > p[i2]) i2 = i;
        const float denom = p[i1] + p[i2];
        for (int i = 0; i < Ec; ++i) {
            const bool is = (i == i1) || (i == i2);
            probs[b * Ec + i] = p[i];
            sel[b * Ec + i]   = is ? 1.0f : 0.0f;
            gate[b * Ec + i]  = is ? (p[i] / denom) : 0.0f;
        }
    }
}

// ---------------- lb partial: lb_acc += E * sum_e mean(probs)*mean(sel) ----------------
__global__ void lb_kernel(const float* __restrict__ probs, const float* __restrict__ sel,
                          float* __restrict__ lb_acc) {
    __shared__ float sp[Ec], ss[Ec];
    const int t = threadIdx.x;
    if (t < Ec) {
        float a = 0.0f, c = 0.0f;
        for (int b = 0; b < Bc; ++b) { a += probs[b * Ec + t]; c += sel[b * Ec + t]; }
        sp[t] = a; ss[t] = c;
    }
    __syncthreads();
    if (t == 0) {
        float s = 0.0f;
        for (int i = 0; i < Ec; ++i) s += (sp[i] / (float)Bc) * (ss[i] / (float)Bc);
        *lb_acc += (float)Ec * s;
    }
}

__global__ void lb_copy_kernel(const float* __restrict__ lb_acc, float* __restrict__ out_lb) {
    if (threadIdx.x == 0 && blockIdx.x == 0) out_lb[0] = *lb_acc;
}

// ---------------- GEMM1: s_new = sigmoid(A)*s + h @ Bin[e]  (per layer) ----------------
// one wave computes one 16x16 tile of (B x DS) for one expert; K = D = 512
__global__ void gemm1_kernel(const float* __restrict__ h,          // (B,D)
                             const float* __restrict__ Bin_l,      // (E,D,DS)
                             const float* __restrict__ states_l,   // (E,B,DS)
                             const float* __restrict__ A_l,        // (E,DS)
                             const float* __restrict__ sel,        // (B,E)
                             float* __restrict__ snew,             // (E,B,DS)
                             float* __restrict__ out_states_l) {   // (E,B,DS)
    const int lane = threadIdx.x & 31;
    const int wave = threadIdx.x >> 5;
    const int w = blockIdx.x * 8 + wave;       // 0..1023
    const int e   = w >> 7;                    // expert
    const int rem = w & 127;
    const int bbase = (rem >> 3) * 16;         // 16 B-tiles
    const int nbase = (rem & 7) * 16;          // 8 DS-tiles

    const int m  = lane & 15;                  // A row / B col
    const int kh = (lane >> 4) * 2;            // K sub-offset {0,2}
    const float* __restrict__ Arow = h + (bbase + m) * Dc;
    const float* __restrict__ Bp   = Bin_l + e * Dc * DSc;

    v8f acc = {};
    for (int k = 0; k < Dc; k += 4) {
        const float2 av = *(const float2*)(Arow + k + kh);
        v2f a; a.x = av.x; a.y = av.y;
        v2f b;
        b.x = Bp[(k + kh)     * DSc + nbase + m];
        b.y = Bp[(k + kh + 1) * DSc + nbase + m];
        acc = __builtin_amdgcn_wmma_f32_16x16x4_f32(false, a, false, b, (short)0, acc,
                                                    false, false);
    }
    const int colhalf = lane >> 4;
    const int ds = nbase + m;
    const float dec = 1.0f / (1.0f + __expf(-A_l[e * DSc + ds]));
#pragma unroll
    for (int j = 0; j < 8; ++j) {
        const int b = bbase + j + colhalf * 8;
        const int idx = (e * Bc + b) * DSc + ds;
        const float sold = states_l[idx];
        const float sv = dec * sold + acc[j];
        snew[idx] = sv;
        out_states_l[idx] = (sel[b * Ec + e] > 0.5f) ? sv : sold;
    }
}

// ---------------- GEMM2: h += sum_e gate[b,e]*(s_new[e]@Cout[e] + h*Dsk[e]) ----------------
// one wave computes one 16x16 tile of (B x D); loops experts; K = DS = 128
__global__ void gemm2_kernel(float* __restrict__ h,               // (B,D) in/out
                             const float* __restrict__ snew,      // (E,B,DS)
                             const float* __restrict__ Cout_l,    // (E,DS,D)
                             const float* __restrict__ Dsk_l,     // (E,D)
                             const float* __restrict__ gate) {    // (B,E)
    const int lane = threadIdx.x & 31;
    const int wave = threadIdx.x >> 5;
    const int w = blockIdx.x * 8 + wave;       // 0..511
    const int bbase = (w >> 5) * 16;           // 16 B-tiles
    const int dbase = (w & 31) * 16;           // 32 D-tiles

    const int m  = lane & 15;
    const int kh = (lane >> 4) * 2;
    const int colhalf = lane >> 4;
    const int dcol = dbase + m;

    float outacc[8] = {0.f, 0.f, 0.f, 0.f, 0.f, 0.f, 0.f, 0.f};
    for (int e = 0; e < Ec; ++e) {
        const float* __restrict__ Ap = snew + (e * Bc + bbase + m) * DSc;
        const float* __restrict__ Bp = Cout_l + e * DSc * Dc;
        v8f acc = {};
        for (int k = 0; k < DSc; k += 4) {
            const float2 av = *(const float2*)(Ap + k + kh);
            v2f a; a.x = av.x; a.y = av.y;
            v2f b;
            b.x = Bp[(k + kh)     * Dc + dcol];
            b.y = Bp[(k + kh + 1) * Dc + dcol];
            acc = __builtin_amdgcn_wmma_f32_16x16x4_f32(false, a, false, b, (short)0, acc,
                                                        false, false);
        }
        const float dk = Dsk_l[e * Dc + dcol];
#pragma unroll
        for (int j = 0; j < 8; ++j) {
            const int b = bbase + j + colhalf * 8;
            const float y = acc[j] + h[b * Dc + dcol] * dk;
            outacc[j] += gate[b * Ec + e] * y;
        }
    }
#pragma unroll
    for (int j = 0; j < 8; ++j) {
        const int b = bbase + j + colhalf * 8;
        h[b * Dc + dcol] += outacc[j];
    }
}

// ---------------- q = h @ Wq + bq ----------------
__global__ void qproj_kernel(const float* __restrict__ h,    // (B,D)
                             const float* __restrict__ Wq,   // (D,D)
                             const float* __restrict__ bq,   // (D)
                             float* __restrict__ q) {        // (B,D)
    const int lane = threadIdx.x & 31;
    const int wave = threadIdx.x >> 5;
    const int w = blockIdx.x * 8 + wave;       // 0..511
    const int bbase = (w >> 5) * 16;
    const int dbase = (w & 31) * 16;
    const int m  = lane & 15;
    const int kh = (lane >> 4) * 2;
    const int colhalf = lane >> 4;
    const int dcol = dbase + m;

    const float* __restrict__ Arow = h + (bbase + m) * Dc;
    v8f acc = {};
    for (int k = 0; k < Dc; k += 4) {
        const float2 av = *(const float2*)(Arow + k + kh);
        v2f a; a.x = av.x; a.y = av.y;
        v2f b;
        b.x = Wq[(k + kh)     * Dc + dcol];
        b.y = Wq[(k + kh + 1) * Dc + dcol];
        acc = __builtin_amdgcn_wmma_f32_16x16x4_f32(false, a, false, b, (short)0, acc,
                                                    false, false);
    }
    const float bias = bq[dcol];
#pragma unroll
    for (int j = 0; j < 8; ++j) {
        const int b = bbase + j + colhalf * 8;
        q[b * Dc + dcol] = acc[j] + bias;
    }
}

// ---------------- logits[b,n] = q[b]·node_emb[b,n] / sqrt(D) ----------------
// one wave per (b,n): streams 2KB per pair, fully coalesced float4 loads
__global__ void logits_kernel(const float* __restrict__ q,          // (B,D)
                              const float* __restrict__ node_emb,   // (B,N,D)
                              float* __restrict__ out) {            // (B,N)
    const int lane = threadIdx.x & 31;
    const int wave = threadIdx.x >> 5;
    const int b = blockIdx.y;
    const int n = blockIdx.x * 8 + wave;
    const float4* __restrict__ np = (const float4*)(node_emb + ((long)b * Nc + n) * Dc);
    const float4* __restrict__ qp = (const float4*)(q + b * Dc);
    float s = 0.0f;
#pragma unroll
    for (int it = 0; it < 4; ++it) {
        const float4 nv = np[it * 32 + lane];
        const float4 qv = qp[it * 32 + lane];
        s += nv.x * qv.x + nv.y * qv.y + nv.z * qv.z + nv.w * qv.w;
    }
    for (int off = 16; off; off >>= 1) s += __shfl_xor(s, off, 32);
    if (lane == 0) out[b * Nc + n] = s * 0.04419417382415922f; // 1/sqrt(512)
}

extern "C" void kernel_launch(void* const* d_in, const int* in_sizes, int n_in,
                              void* d_out, int out_size, void* d_ws, size_t ws_size,
                              hipStream_t stream) {
    const float* token    = (const float*)d_in[0];   // (B,1,D)
    const float* node_emb = (const float*)d_in[1];   // (B,N,D)
    const float* states   = (const float*)d_in[2];   // (L,E,B,DS)
    const float* Wg       = (const float*)d_in[3];   // (L,D,E)
    const float* bg       = (const float*)d_in[4];   // (L,E)
    const float* A        = (const float*)d_in[5];   // (L,E,DS)
    const float* Bin      = (const float*)d_in[6];   // (L,E,D,DS)
    const float* Cout     = (const float*)d_in[7];   // (L,E,DS,D)
    const float* Dsk      = (const float*)d_in[8];   // (L,E,D)
    const float* Wq       = (const float*)d_in[9];   // (D,D)
    const float* bq       = (const float*)d_in[10];  // (D)

    float* out_logits = (float*)d_out;                       // B*N
    float* out_states = out_logits + Bc * Nc;                // L*E*B*DS
    float* out_lb     = out_states + Lc * Ec * Bc * DSc;     // 1

    // workspace layout (floats)
    float* ws   = (float*)d_ws;
    float* h    = ws;                         // B*D      = 131072
    float* q    = h + Bc * Dc;                // B*D      = 131072
    float* snew = q + Bc * Dc;                // E*B*DS   = 262144
    float* probs = snew + Ec * Bc * DSc;      // B*E
    float* gate  = probs + Bc * Ec;           // B*E
    float* sel   = gate + Bc * Ec;            // B*E
    float* lbacc = sel + Bc * Ec;             // 1

    init_kernel<<<(Bc * Dc + 255) / 256, 256, 0, stream>>>(token, h, lbacc);

    for (int l = 0; l < Lc; ++l) {
        const float* Wg_l  = Wg  + l * Dc * Ec;
        const float* bg_l  = bg  + l * Ec;
        const float* A_l   = A   + l * Ec * DSc;
        const float* Bin_l = Bin + (size_t)l * Ec * Dc * DSc;
        const float* Cout_l= Cout+ (size_t)l * Ec * DSc * Dc;
        const float* Dsk_l = Dsk + l * Ec * Dc;
        const float* st_l  = states + (size_t)l * Ec * Bc * DSc;
        float* ost_l       = out_states + (size_t)l * Ec * Bc * DSc;

        gate_kernel<<<Bc, 256, 0, stream>>>(h, Wg_l, bg_l, probs, gate, sel);
        lb_kernel<<<1, 32, 0, stream>>>(probs, sel, lbacc);
        gemm1_kernel<<<128, 256, 0, stream>>>(h, Bin_l, st_l, A_l, sel, snew, ost_l);
        gemm2_kernel<<<64, 256, 0, stream>>>(h, snew, Cout_l, Dsk_l, gate);
    }

    qproj_kernel<<<64, 256, 0, stream>>>(h, Wq, bq, q);
    logits_kernel<<<dim3(Nc / 8, Bc), 256, 0, stream>>>(q, node_emb, out_logits);
    lb_copy_kernel<<<1, 32, 0, stream>>>(lbacc, out_lb);
}